// WHVISquarePow2Matrix_54348516164119
// MI455X (gfx1250) — compile-verified
//
#include <hip/hip_runtime.h>

typedef __attribute__((ext_vector_type(2))) float v2f;
typedef __attribute__((ext_vector_type(8))) float v8f;

#define DIM 4096

// H16 Sylvester entry: (-1)^popc(m&k)
__device__ __forceinline__ float hval(int m, int k) {
    return (__popc(m & k) & 1) ? -1.0f : 1.0f;
}

// One 16x16 tile of "H16 @ M" where M[k][col] = buf[base + k*sK + col*sC].
// Uses 4 chained V_WMMA_F32_16X16X4_F32 (K = 4 per step, fp32-exact).
__device__ __forceinline__ v8f tile_fwht(const float* __restrict__ buf,
                                         int base, int sK, int sC,
                                         const v2f* __restrict__ afrag,
                                         int hf, int l16) {
    v8f acc = {};
#pragma unroll
    for (int c = 0; c < 4; ++c) {
        const int k0 = 4 * c + 2 * hf;
        v2f b;
        b.x = buf[base + (k0    ) * sK + l16 * sC];
        b.y = buf[base + (k0 + 1) * sK + l16 * sC];
        acc = __builtin_amdgcn_wmma_f32_16x16x4_f32(
            /*neg_a=*/false, afrag[c], /*neg_b=*/false, b,
            /*c_mod=*/(short)0, acc, /*reuse_a=*/false, /*reuse_b=*/false);
    }
    return acc;
}

__device__ __forceinline__ void tile_store_lds(float* __restrict__ buf,
                                               int base, int sK, int sC,
                                               const v8f& d, int hf, int l16) {
#pragma unroll
    for (int r = 0; r < 8; ++r)
        buf[base + (r + 8 * hf) * sK + l16 * sC] = d[r];
}

// g = g_mu + softplus(g_rho) * epsilon
__global__ void whvi_g_kernel(const float* __restrict__ g_mu,
                              const float* __restrict__ g_rho,
                              const float* __restrict__ eps,
                              float* __restrict__ g) {
    int i = blockIdx.x * blockDim.x + threadIdx.x;
    if (i < DIM) {
        float sp = log1pf(expf(g_rho[i]));
        g[i] = g_mu[i] + sp * eps[i];
    }
}

// One block (512 threads = 16 waves) per row of x.
// Row viewed as 16x16x16 tensor idx = i*256 + j*16 + k; FWHT = H16 along each axis.
__global__ __launch_bounds__(512) void whvi_main(const float* __restrict__ x,
                                                 const float* __restrict__ s1,
                                                 const float* __restrict__ s2,
                                                 const float* __restrict__ g,
                                                 float* __restrict__ out) {
    __shared__ float buf[DIM];

    const int row  = blockIdx.x;
    const int tid  = threadIdx.x;
    const int t    = tid >> 5;        // wave id == tensor slice / tile id (0..15)
    const int lane = tid & 31;
    const int hf   = lane >> 4;       // lane half
    const int l16  = lane & 15;

    // Stream row in, fuse s2 scaling. Coalesced: stride-512 across threads.
    const float* xr = x + (size_t)row * DIM;
#pragma unroll
    for (int e = 0; e < 8; ++e) {
        int idx = tid + 512 * e;
        buf[idx] = xr[idx] * s2[idx];
    }
    __syncthreads();

    // A fragments = H16 rows, built from popcount parity (no memory).
    v2f afrag[4];
#pragma unroll
    for (int c = 0; c < 4; ++c) {
        const int k0 = 4 * c + 2 * hf;
        afrag[c].x = hval(l16, k0);
        afrag[c].y = hval(l16, k0 + 1);
    }

    v8f d;

    // ---------------- FWHT #1 ----------------
    // axis k: M[k][col] = buf[t*256 + col*16 + k]   (wave-local region, no barrier)
    d = tile_fwht(buf, t * 256, /*sK=*/1, /*sC=*/16, afrag, hf, l16);
    tile_store_lds(buf, t * 256, 1, 16, d, hf, l16);
    // axis j: M[j][col] = buf[t*256 + j*16 + col]   (same wave-local region)
    d = tile_fwht(buf, t * 256, /*sK=*/16, /*sC=*/1, afrag, hf, l16);
    tile_store_lds(buf, t * 256, 16, 1, d, hf, l16);
    __syncthreads();
    // axis i: M[i][col] = buf[i*256 + t*16 + col]   (mixes all slices)
    d = tile_fwht(buf, t * 16, /*sK=*/256, /*sC=*/1, afrag, hf, l16);
#pragma unroll
    for (int r = 0; r < 8; ++r) {           // fuse elementwise g
        int a = t * 16 + (r + 8 * hf) * 256 + l16;
        buf[a] = d[r] * g[a];
    }
    __syncthreads();

    // ---------------- FWHT #2 ----------------
    d = tile_fwht(buf, t * 256, 1, 16, afrag, hf, l16);
    tile_store_lds(buf, t * 256, 1, 16, d, hf, l16);
    d = tile_fwht(buf, t * 256, 16, 1, afrag, hf, l16);
    tile_store_lds(buf, t * 256, 16, 1, d, hf, l16);
    __syncthreads();
    d = tile_fwht(buf, t * 16, 256, 1, afrag, hf, l16);

    // Fuse s1 scaling into the global store.
    float* outr = out + (size_t)row * DIM;
#pragma unroll
    for (int r = 0; r < 8; ++r) {
        int a = t * 16 + (r + 8 * hf) * 256 + l16;
        outr[a] = d[r] * s1[a];
    }
}

extern "C" void kernel_launch(void* const* d_in, const int* in_sizes, int n_in,
                              void* d_out, int out_size, void* d_ws, size_t ws_size,
                              hipStream_t stream) {
    const float* x     = (const float*)d_in[0];
    const float* eps   = (const float*)d_in[1];
    const float* s1    = (const float*)d_in[2];
    const float* s2    = (const float*)d_in[3];
    const float* g_mu  = (const float*)d_in[4];
    const float* g_rho = (const float*)d_in[5];
    float* out = (float*)d_out;
    float* g   = (float*)d_ws;   // DIM floats of scratch

    const int rows = in_sizes[0] / DIM;   // 2048

    whvi_g_kernel<<<(DIM + 255) / 256, 256, 0, stream>>>(g_mu, g_rho, eps, g);
    whvi_main<<<rows, 512, 0, stream>>>(x, s1, s2, g, out);
}